// GATLayer_72069551227455
// MI455X (gfx1250) — compile-verified
//
#include <hip/hip_runtime.h>
#include <cstdint>
#include <cstddef>

#define N_NODES 8192
#define E_EDGES 262144
#define IN_FEATS 256
#define OUT_FEATS 64
#define H_HEADS 2
#define SLOPE 0.2f

typedef float v2f __attribute__((ext_vector_type(2)));
typedef float v8f __attribute__((ext_vector_type(8)));

// ---------------------------------------------------------------------------
// Kernel 1: xh[h,n,:] = x[n,:] @ W[h]  via V_WMMA_F32_16X16X4_F32 (fp32 exact)
// One wave computes a 16-row stripe across both heads and all 4 N-tiles
// (8 accumulators), so each A-tile load is reused by 8 WMMAs.
// ---------------------------------------------------------------------------
__global__ __launch_bounds__(256) void gat_xh_wmma(const float* __restrict__ x,
                                                   const float* __restrict__ W,
                                                   float* __restrict__ xh) {
  const int lane = threadIdx.x & 31;
  const int wid  = blockIdx.x * (blockDim.x >> 5) + (threadIdx.x >> 5);
  if (wid >= N_NODES / 16) return;

  const int m0   = wid * 16;
  const int row  = m0 + (lane & 15);       // A: lanes 0-15 and 16-31 both hold M=0..15
  const int ksub = (lane >> 4) << 1;       // A: v0/v1 = K0/K1 (lo lanes), K2/K3 (hi lanes)
  const int col  = lane & 15;              // B/D: N = lane & 15

  v8f acc[H_HEADS][4];
#pragma unroll
  for (int h = 0; h < H_HEADS; ++h)
#pragma unroll
    for (int t = 0; t < 4; ++t) acc[h][t] = (v8f){0.f,0.f,0.f,0.f,0.f,0.f,0.f,0.f};

  for (int k = 0; k < IN_FEATS; k += 4) {
    const int kk = k + ksub;                                // even -> 8B aligned
    v2f a = *(const v2f*)(x + (size_t)row * IN_FEATS + kk); // A 16x4 tile
#pragma unroll
    for (int h = 0; h < H_HEADS; ++h) {
      const float* Wh = W + (size_t)h * IN_FEATS * OUT_FEATS;
#pragma unroll
      for (int t = 0; t < 4; ++t) {
        v2f b;                                              // B 4x16 tile, cols t*16..t*16+15
        b.x = Wh[(size_t)kk * OUT_FEATS + t * 16 + col];
        b.y = Wh[(size_t)(kk + 1) * OUT_FEATS + t * 16 + col];
        acc[h][t] = __builtin_amdgcn_wmma_f32_16x16x4_f32(
            false, a, false, b, (short)0, acc[h][t], false, false);
      }
    }
  }

  // D layout: VGPR v -> row m0 + v + 8*(lane>=16), col = lane&15
  const int rbase = m0 + ((lane >> 4) << 3);
#pragma unroll
  for (int h = 0; h < H_HEADS; ++h)
#pragma unroll
    for (int t = 0; t < 4; ++t)
#pragma unroll
      for (int v = 0; v < 8; ++v)
        xh[((size_t)h * N_NODES + rbase + v) * OUT_FEATS + t * 16 + col] = acc[h][t][v];
}

// ---------------------------------------------------------------------------
// Kernel 2: s1[h,n] = xh[h,n,:].a1[h], s2 likewise. One wave per (h,n) row,
// wave32 shuffle reduction.
// ---------------------------------------------------------------------------
__global__ __launch_bounds__(256) void gat_scores(const float* __restrict__ xh,
                                                  const float* __restrict__ a1,
                                                  const float* __restrict__ a2,
                                                  float* __restrict__ s1,
                                                  float* __restrict__ s2) {
  const int lane = threadIdx.x & 31;
  const int hn   = blockIdx.x * (blockDim.x >> 5) + (threadIdx.x >> 5);
  if (hn >= H_HEADS * N_NODES) return;
  const int h = hn / N_NODES;

  const float* r = xh + (size_t)hn * OUT_FEATS;
  float x0 = r[lane], x1 = r[lane + 32];
  float p1 = x0 * a1[h * OUT_FEATS + lane] + x1 * a1[h * OUT_FEATS + lane + 32];
  float p2 = x0 * a2[h * OUT_FEATS + lane] + x1 * a2[h * OUT_FEATS + lane + 32];
#pragma unroll
  for (int m = 16; m >= 1; m >>= 1) {
    p1 += __shfl_xor(p1, m, 32);
    p2 += __shfl_xor(p2, m, 32);
  }
  if (lane == 0) { s1[hn] = p1; s2[hn] = p2; }
}

// ---------------------------------------------------------------------------
// Kernel 3: column means of xh (fallback for zero-degree rows = uniform softmax)
// ---------------------------------------------------------------------------
__global__ void gat_colmean(const float* __restrict__ xh, float* __restrict__ cm) {
  const int t = threadIdx.x;
  if (t >= H_HEADS * OUT_FEATS) return;
  float s = 0.f;
  for (int n = 0; n < N_NODES; ++n)
    s += xh[(size_t)(t >> 6) * N_NODES * OUT_FEATS + (size_t)n * OUT_FEATS + (t & 63)];
  cm[t] = s * (1.0f / N_NODES);
}

// ---------------------------------------------------------------------------
// Kernel 4: edge-parallel masked softmax accumulation. One wave per edge.
// Dedup duplicate COO edges via bitmap test-and-set (reference uses adj>0).
// w = exp(leaky(s1[src]+s2[dst])); accumulate denom and w*xh[dst] into src row.
// ---------------------------------------------------------------------------
__global__ __launch_bounds__(256) void gat_edges(const long long* __restrict__ ei,
                                                 const float* __restrict__ s1,
                                                 const float* __restrict__ s2,
                                                 const float* __restrict__ xh,
                                                 float* __restrict__ denom,
                                                 float* __restrict__ accum,
                                                 unsigned* __restrict__ bitmap) {
  const int lane = threadIdx.x & 31;
  const int e    = blockIdx.x * (blockDim.x >> 5) + (threadIdx.x >> 5);
  if (e >= E_EDGES) return;

  const int src = (int)ei[e];
  const int dst = (int)ei[E_EDGES + e];

  unsigned dup = 0;
  if (lane == 0) {
    const unsigned long long idx = (unsigned long long)src * N_NODES + (unsigned)dst;
    unsigned old = atomicOr(&bitmap[idx >> 5], 1u << (idx & 31u));
    dup = (old >> (idx & 31u)) & 1u;
  }
  dup = __shfl(dup, 0, 32);
  if (dup) return;  // duplicate edge collapses to one in the reference mask

  float w[H_HEADS];
#pragma unroll
  for (int h = 0; h < H_HEADS; ++h) {
    float sc = s1[h * N_NODES + src] + s2[h * N_NODES + dst];
    sc = (sc >= 0.f) ? sc : SLOPE * sc;  // leaky_relu
    w[h] = __expf(sc);                   // |sc| << 88: safe without max-shift
  }
  if (lane < H_HEADS) atomicAdd(&denom[lane * N_NODES + src], w[lane]);

#pragma unroll
  for (int h = 0; h < H_HEADS; ++h) {
    const float* xr = xh    + ((size_t)h * N_NODES + dst) * OUT_FEATS;
    float*       ar = accum + ((size_t)h * N_NODES + src) * OUT_FEATS;
    atomicAdd(&ar[lane],      w[h] * xr[lane]);
    atomicAdd(&ar[lane + 32], w[h] * xr[lane + 32]);
  }
}

// ---------------------------------------------------------------------------
// Kernel 5: out[n, h*64+o] = accum / denom (or uniform mean if no neighbors)
// ---------------------------------------------------------------------------
__global__ void gat_finalize(const float* __restrict__ accum,
                             const float* __restrict__ denom,
                             const float* __restrict__ cm,
                             float* __restrict__ out) {
  const int tid = blockIdx.x * blockDim.x + threadIdx.x;
  if (tid >= N_NODES * H_HEADS * OUT_FEATS) return;
  const int n = tid >> 7;
  const int c = tid & 127;
  const int h = c >> 6;
  const int o = c & 63;
  const float d = denom[h * N_NODES + n];
  out[tid] = (d > 0.f) ? accum[((size_t)h * N_NODES + n) * OUT_FEATS + o] / d : cm[c];
}

// ---------------------------------------------------------------------------
extern "C" void kernel_launch(void* const* d_in, const int* in_sizes, int n_in,
                              void* d_out, int out_size, void* d_ws, size_t ws_size,
                              hipStream_t stream) {
  (void)in_sizes; (void)n_in; (void)out_size;
  const float*     x  = (const float*)d_in[0];
  const long long* ei = (const long long*)d_in[1];  // int64 [2, E]
  const float*     W  = (const float*)d_in[2];      // [H, 256, 64]
  const float*     a1 = (const float*)d_in[3];
  const float*     a2 = (const float*)d_in[4];
  float*           out = (float*)d_out;

  const size_t ACC = (size_t)H_HEADS * N_NODES * OUT_FEATS;  // floats (4 MB)
  const size_t DEN = (size_t)H_HEADS * N_NODES;              // floats (64 KB)
  const size_t BMP = ((size_t)N_NODES * N_NODES) / 32;       // uints  (8 MB)
  const size_t XH  = ACC;                                    // floats (4 MB)
  const size_t S   = DEN;                                    // floats

  float*    accum  = (float*)d_ws;
  float*    denom  = accum + ACC;
  unsigned* bitmap = (unsigned*)(denom + DEN);
  float*    xh     = (float*)(bitmap + BMP);
  float*    s1     = xh + XH;
  float*    s2     = s1 + S;
  float*    cm     = s2 + S;

  const size_t need = (ACC + DEN + BMP + XH + 2 * S + H_HEADS * OUT_FEATS) * 4;
  if (ws_size < need) return;

  // zero accum + denom + bitmap in one shot (graph-capture safe)
  hipMemsetAsync(d_ws, 0, (ACC + DEN + BMP) * 4, stream);

  gat_xh_wmma <<<(N_NODES / 16 + 7) / 8, 256, 0, stream>>>(x, W, xh);
  gat_scores  <<<(H_HEADS * N_NODES + 7) / 8, 256, 0, stream>>>(xh, a1, a2, s1, s2);
  gat_colmean <<<1, 128, 0, stream>>>(xh, cm);
  gat_edges   <<<E_EDGES / 8, 256, 0, stream>>>(ei, s1, s2, xh, denom, accum, bitmap);
  gat_finalize<<<(N_NODES * H_HEADS * OUT_FEATS + 255) / 256, 256, 0, stream>>>(accum, denom, cm, out);
}